// MotoneuronNetwork_7086696038946
// MI455X (gfx1250) — compile-verified
//
#include <hip/hip_runtime.h>
#include <hip/hip_bf16.h>

// ---------------- model constants ----------------
#define NN      1024
#define BB      8
#define BPAD    16                      // batch padded to WMMA N=16
#define TT      512
#define DT      0.05f
#define THRESH  (-20.0f)
#define V_RESET (-60.0f)
#define DECAY   0.97530991202833262f   // exp(-DT/TAU_SYN) = exp(-0.025)
#define W_SCALE (1.0f/64.0f)

#define NBLOCKS 64                      // one 16-row output tile per block

typedef __attribute__((ext_vector_type(2))) float v2f;
typedef __attribute__((ext_vector_type(8))) float v8f;

__device__ __forceinline__ float xoexpm1(float x) {
    // numerically safe x / (exp(x) - 1)
    float ax   = fabsf(x);
    float safe = (ax < 1e-6f) ? 1.0f : x;
    float r    = safe / (__expf(safe) - 1.0f);
    return (ax < 1e-6f) ? (1.0f - 0.5f * x) : r;
}

// zero barrier counters + both padded spike buffers (rows 8..15 stay 0 forever)
__global__ void pr_init(unsigned* bar, float* spk_buf) {
    int g = blockIdx.x * blockDim.x + threadIdx.x;
    if (g < 2) bar[g] = 0u;
    for (int i = g; i < 2 * BPAD * NN; i += gridDim.x * blockDim.x)
        spk_buf[i] = 0.0f;
}

extern "C" __global__ __launch_bounds__(256, 1)
void pr_persistent(const float* __restrict__ y0,
                   const float* __restrict__ w,
                   const float* __restrict__ I_ext,
                   const int*   __restrict__ network,
                   float* __restrict__ vs_out,
                   float* __restrict__ spk_out,
                   float* __restrict__ spk_buf,   // 2 * BPAD*NN floats (ping-pong)
                   unsigned* __restrict__ bar)
{
    __shared__ float lds_red[8 * 256];   // per-wave partial C tiles
    __shared__ float isyn_lds[128];      // I_syn for this block's (b, n) pairs

    const int tid  = threadIdx.x;
    const int bi   = blockIdx.x;         // 0..63 -> output rows [16*bi, 16*bi+16)
    const int i0   = bi * 16;
    const int lane = tid & 31;
    const int wave = tid >> 5;           // 0..7, owns K range [wave*128, +128)
    const int hi   = lane >> 4;          // 0/1 half-wave
    const int ln   = lane & 15;

    // ---- load this wave's A-slice of w_eff into WMMA-layout registers (once) ----
    // A layout (16x4 fp32): lanes0-15 M=0..15 {K0,K1}; lanes16-31 M=0..15 {K2,K3}
    // FULL unroll => constant indices => a_reg promoted to VGPRs (no scratch).
    v2f a_reg[32];
    {
        const int row   = i0 + ln;
        const int kbase = wave * 128 + 2 * hi;
#pragma unroll
        for (int kc = 0; kc < 32; ++kc) {
            int idx = row * NN + kbase + 4 * kc;
            float w0 = w[idx], w1 = w[idx + 1];
            int   m0 = network[idx], m1 = network[idx + 1];
            a_reg[kc].x = m0 ? 0.0f : w0 * W_SCALE;   // network==True -> zero
            a_reg[kc].y = m1 ? 0.0f : w1 * W_SCALE;
        }
    }

    // ---- persistent neuron state: threads 0..127 own (b = tid>>4, n = i0 + (tid&15)) ----
    float Vs = 0.f, Vd = 0.f, Ca = 0.f, hv = 0.f, nv = 0.f, sv = 0.f, cv = 0.f, qv = 0.f;
    float Iext = 0.f;
    int b = 0, nrn = 0;
    if (tid < 128) {
        b   = tid >> 4;
        nrn = i0 + (tid & 15);
        const float* yp = y0 + ((size_t)b * NN + nrn) * 8;
        Vs = yp[0]; Vd = yp[1]; Ca = yp[2]; hv = yp[3];
        nv = yp[4]; sv = yp[5]; cv = yp[6]; qv = yp[7];
        Iext = I_ext[b * NN + nrn];
        isyn_lds[tid] = 0.0f;            // I_syn0 = 0
    }
    __syncthreads();

    unsigned gen_target = 0;

    for (int t = 0; t < TT; ++t) {
        float* spk_cur = spk_buf + (size_t)(t & 1) * (BPAD * NN);

        // ---------- phase 1: pointwise Pinsky-Rinzel Euler step + spike ----------
        if (tid < 128) {
            float Isyn = isyn_lds[tid];

            float am = 1.28f * xoexpm1((-46.9f - Vs) * 0.25f);
            float bm = 1.4f  * xoexpm1((Vs + 19.9f) * 0.2f);
            float m_inf = am / (am + bm);
            float ah = 0.128f * __expf((-43.0f - Vs) * (1.0f / 18.0f));
            float bh = 4.0f / (1.0f + __expf((-20.0f - Vs) * 0.2f));
            float an = 0.08f * xoexpm1((-24.9f - Vs) * 0.2f);
            float bn = 0.25f * __expf(-1.0f - 0.025f * Vs);
            float as_ = 1.6f / (1.0f + __expf(-0.072f * (Vd - 5.0f)));
            float bs = 0.1f * xoexpm1((Vd + 8.9f) * 0.2f);

            bool  low     = (Vd <= -10.0f);
            float ac_low  = __expf((Vd + 50.0f) * (1.0f / 11.0f)
                                   - (Vd + 53.5f) * (1.0f / 27.0f)) * (1.0f / 18.975f);
            float ac_high = 2.0f * __expf((-53.5f - Vd) * (1.0f / 27.0f));
            float ac = low ? ac_low : ac_high;
            float bc = low ? (ac_high - ac_low) : 0.0f;

            float aq  = fminf(0.00002f * Ca, 0.01f);
            float chi = fminf(Ca * (1.0f / 250.0f), 1.0f);

            float I_Na  = 30.0f * m_inf * m_inf * hv * (Vs - 60.0f);
            float I_DR  = 15.0f * nv * (Vs + 75.0f);
            float I_Ca  = 10.0f * sv * sv * (Vd - 80.0f);
            float I_AHP = 0.8f  * qv * (Vd + 75.0f);
            float I_C   = 15.0f * cv * chi * (Vd + 75.0f);

            float dVs = (-0.1f * (Vs + 60.0f) - I_Na - I_DR
                         + 4.2f * (Vd - Vs) + 2.0f * Iext) * (1.0f / 3.0f);
            float dVd = (-0.1f * (Vd + 60.0f) - I_Ca - I_AHP - I_C
                         + 4.2f * (Vs - Vd) + 2.0f * Isyn) * (1.0f / 3.0f);
            float dCa = -0.13f * I_Ca - 0.075f * Ca;
            float dh  = ah  * (1.0f - hv) - bh * hv;
            float dn  = an  * (1.0f - nv) - bn * nv;
            float ds_ = as_ * (1.0f - sv) - bs * sv;
            float dc  = ac  * (1.0f - cv) - bc * cv;
            float dq  = aq  * (1.0f - qv) - 0.001f * qv;

            float Vs_old = Vs;
            Vs += DT * dVs; Vd += DT * dVd; Ca += DT * dCa; hv += DT * dh;
            nv += DT * dn;  sv += DT * ds_; cv += DT * dc;  qv += DT * dq;

            float spike = (Vs >= THRESH && Vs_old < THRESH) ? 1.0f : 0.0f;
            if (spike != 0.0f) Vs = V_RESET;

            size_t ofs = (size_t)b * TT * NN + (size_t)t * NN + nrn;
            vs_out[ofs]  = Vs;
            spk_out[ofs] = spike;
            __hip_atomic_store(&spk_cur[b * NN + nrn], spike,
                               __ATOMIC_RELAXED, __HIP_MEMORY_SCOPE_AGENT);
        }

        // ---------- device-wide barrier (spikes visible to all WGPs) ----------
        __threadfence();
        __syncthreads();
        ++gen_target;
        if (tid == 0) {
            unsigned old = __hip_atomic_fetch_add(&bar[0], 1u,
                               __ATOMIC_ACQ_REL, __HIP_MEMORY_SCOPE_AGENT);
            if (old == NBLOCKS - 1) {
                __hip_atomic_store(&bar[0], 0u,
                                   __ATOMIC_RELAXED, __HIP_MEMORY_SCOPE_AGENT);
                __hip_atomic_store(&bar[1], gen_target,
                                   __ATOMIC_RELEASE, __HIP_MEMORY_SCOPE_AGENT);
            } else {
                while (__hip_atomic_load(&bar[1], __ATOMIC_ACQUIRE,
                                         __HIP_MEMORY_SCOPE_AGENT) < gen_target) {
                    __builtin_amdgcn_s_sleep(2);
                }
            }
        }
        __syncthreads();

        // ---------- phase 2: WMMA f32 16x16x4 tile, K split across 8 waves ----------
        // D[M][N] += A[M][K] * B[K][N];  M = neuron row, N = batch (8 real + 8 zero-pad)
        // B layout (4x16 fp32): lanes0-15 N=0..15 {K0,K1}; lanes16-31 N=0..15 {K2,K3}
        const int kb = wave * 128 + 2 * hi;
        v8f acc = {};
#pragma unroll
        for (int kc = 0; kc < 32; ++kc) {
            int k = kb + 4 * kc;
            v2f bv;
            bv.x = __hip_atomic_load(&spk_cur[ln * NN + k],
                                     __ATOMIC_RELAXED, __HIP_MEMORY_SCOPE_AGENT);
            bv.y = __hip_atomic_load(&spk_cur[ln * NN + k + 1],
                                     __ATOMIC_RELAXED, __HIP_MEMORY_SCOPE_AGENT);
            acc = __builtin_amdgcn_wmma_f32_16x16x4_f32(
                      false, a_reg[kc], false, bv, (short)0, acc, false, false);
        }

        // ---------- phase 3: reduce 8 wave-partials in LDS, update I_syn ----------
#pragma unroll
        for (int r = 0; r < 8; ++r)
            lds_red[wave * 256 + lane * 8 + r] = acc[r];
        __syncthreads();
        {
            float ssum = 0.0f;
#pragma unroll
            for (int wv = 0; wv < 8; ++wv)
                ssum += lds_red[wv * 256 + tid];
            int rl  = tid & 7;          // accumulator VGPR index
            int lnn = tid >> 3;         // lane
            int M   = rl + 8 * (lnn >> 4);
            int bb  = lnn & 15;         // N dim = batch
            if (bb < 8) {
                int sl = bb * 16 + M;   // matches state-thread tid = b*16 + m
                isyn_lds[sl] = isyn_lds[sl] * DECAY + ssum;
            }
        }
        __syncthreads();
    }
}

extern "C" void kernel_launch(void* const* d_in, const int* in_sizes, int n_in,
                              void* d_out, int out_size, void* d_ws, size_t ws_size,
                              hipStream_t stream) {
    (void)in_sizes; (void)n_in; (void)out_size; (void)ws_size;
    const float* y0      = (const float*)d_in[0];   // [B,N,8]
    const float* w       = (const float*)d_in[1];   // [N,N]
    const float* I_ext   = (const float*)d_in[2];   // [B,N]
    const int*   network = (const int*)d_in[3];     // [N,N] bool->int
    // d_in[4] = T (fixed 512)

    float* out     = (float*)d_out;
    float* vs_out  = out;                                   // [B,T,N]
    float* spk_out = out + (size_t)BB * TT * NN;            // [B,T,N]

    unsigned* bar     = (unsigned*)d_ws;                    // 2 counters
    float*    spk_buf = (float*)((char*)d_ws + 256);        // 2 * BPAD*NN floats

    pr_init<<<32, 256, 0, stream>>>(bar, spk_buf);
    pr_persistent<<<NBLOCKS, 256, 0, stream>>>(y0, w, I_ext, network,
                                               vs_out, spk_out, spk_buf, bar);
}